// GraphAttentionLayer_86500641341504
// MI455X (gfx1250) — compile-verified
//
#include <hip/hip_runtime.h>
#include <hip/hip_bf16.h>

// ---- problem constants (match reference) ----
#define NNODES 50000
#define KNBR   12
#define FDIM   128      // F_IN == F_OUT == 128
#define ALPHA  0.2f

#define ROWS_PB 80      // rows per block (5 x 16-row WMMA tiles); 50000/80 = 625
#define RTILES  5

typedef __attribute__((ext_vector_type(2))) float v2f;
typedef __attribute__((ext_vector_type(8))) float v8f;

// -----------------------------------------------------------------------------
// Kernel 0: one-time transpose of both weight matrices into workspace so that
// the WMMA B-fragment (fixed n, consecutive k) becomes a single aligned b64.
// -----------------------------------------------------------------------------
__global__ __launch_bounds__(256)
void transpose_w_kernel(const float* __restrict__ Ws,
                        const float* __restrict__ Wn,
                        float* __restrict__ WtS,
                        float* __restrict__ WtN)
{
    const int idx = blockIdx.x * 256 + threadIdx.x;   // 0..16383
    const int k = idx >> 7;
    const int n = idx & 127;
    WtS[n * FDIM + k] = Ws[k * FDIM + n];
    WtN[n * FDIM + k] = Wn[k * FDIM + n];
}

// -----------------------------------------------------------------------------
// Kernel 1: fused dual GEMM (act_self -> out, h -> ws) + per-node attention
// dots s_in = h@a[:128], s_out = h@a[128:].
// Block = 256 threads = 8 waves, 80 rows per block. Wave w owns column tile
// n0 = 16*w for BOTH GEMMs and all 5 row tiles, so every B-fragment load is
// reused by 5 WMMAs.
// -----------------------------------------------------------------------------
__global__ __launch_bounds__(256)
void gat_gemm_kernel(const float* __restrict__ x,
                     const float* __restrict__ WtS,    // transposed [n][k]
                     const float* __restrict__ WtN,    // transposed [n][k]
                     const float* __restrict__ avec,   // [256]
                     float* __restrict__ out,          // [N,128] act_self
                     float* __restrict__ h,            // [N,128] scratch
                     float* __restrict__ s_in,         // [N]
                     float* __restrict__ s_out)        // [N]
{
    __shared__ float lds_x[ROWS_PB * FDIM];   // 40 KB A tiles
    __shared__ float lds_h[ROWS_PB * FDIM];   // 40 KB h tiles (for s dots)

    const int tid  = threadIdx.x;
    const int wave = tid >> 5;           // 0..7
    const int lane = tid & 31;
    const int r0   = blockIdx.x * ROWS_PB;

    // cooperative stage of the 80x128 x-tile into LDS (float4, 10 per thread)
    {
        const float4* xsrc = (const float4*)(x + (size_t)r0 * FDIM);
        float4*       xdst = (float4*)lds_x;
        #pragma unroll
        for (int i = 0; i < 10; ++i)
            xdst[tid + i * 256] = xsrc[tid + i * 256];
    }
    __syncthreads();

    // WMMA 16x16x4 f32 fragment indexing (ISA 7.12.2):
    //   lane L: m/n = L%16, k = 2*(L/16) + j  (j = vgpr 0/1)
    const int m    = lane & 15;
    const int kgrp = (lane >> 4) << 1;   // 0 or 2
    const int n0   = wave * 16;

    v8f accS[RTILES];
    v8f accN[RTILES];
    #pragma unroll
    for (int t = 0; t < RTILES; ++t) { accS[t] = (v8f){}; accN[t] = (v8f){}; }

    const float* wcS = WtS + (size_t)(n0 + m) * FDIM;  // column n0+m of W_self
    const float* wcN = WtN + (size_t)(n0 + m) * FDIM;  // column n0+m of W_ngb

    #pragma unroll 8
    for (int k0 = 0; k0 < FDIM; k0 += 4) {
        const int k = k0 + kgrp;                       // even -> 8B aligned
        const v2f bs = *(const v2f*)(wcS + k);         // one b64 per matrix
        const v2f bn = *(const v2f*)(wcN + k);
        #pragma unroll
        for (int t = 0; t < RTILES; ++t) {
            const v2f a = *(const v2f*)&lds_x[(t * 16 + m) * FDIM + k];
            accS[t] = __builtin_amdgcn_wmma_f32_16x16x4_f32(
                          false, a, false, bs, (short)0, accS[t], false, false);
            accN[t] = __builtin_amdgcn_wmma_f32_16x16x4_f32(
                          false, a, false, bn, (short)0, accN[t], false, false);
        }
    }

    // C/D layout: VGPR r -> row = r + 8*(lane/16), col = lane%16
    const int mrow = (lane >> 4) << 3;   // 0 or 8
    #pragma unroll
    for (int t = 0; t < RTILES; ++t) {
        #pragma unroll
        for (int r = 0; r < 8; ++r) {
            const int lrow = t * 16 + r + mrow;        // 0..79 within block
            const int row  = r0 + lrow;
            const int col  = n0 + m;
            out[(size_t)row * FDIM + col] = accS[t][r];   // act_self
            h  [(size_t)row * FDIM + col] = accN[t][r];   // h for aggregation
            lds_h[lrow * FDIM + col]      = accN[t][r];   // h for s dots
        }
    }
    __syncthreads();

    // s_in / s_out: 80 rows / 8 waves -> 10 rows per wave
    #pragma unroll
    for (int rr = 0; rr < 10; ++rr) {
        const int lrow = wave * 10 + rr;
        float a0 = 0.f, a1 = 0.f;
        #pragma unroll
        for (int c = lane; c < FDIM; c += 32) {
            const float hv = lds_h[lrow * FDIM + c];
            a0 += hv * avec[c];
            a1 += hv * avec[FDIM + c];
        }
        #pragma unroll
        for (int off = 16; off > 0; off >>= 1) {
            a0 += __shfl_down(a0, off, 32);
            a1 += __shfl_down(a1, off, 32);
        }
        if (lane == 0) {
            s_in [r0 + lrow] = a0;
            s_out[r0 + lrow] = a1;
        }
    }
}

// -----------------------------------------------------------------------------
// Kernel 2: per-node LeakyReLU + softmax over the K=12 contiguous edges, then
// attention-weighted gather-sum of h[col] rows, + act_self (already in out)
// + bias. One 128-thread block per node -> h-row gathers are coalesced and
// L2-resident (h is 25.6 MB << 192 MB L2).
// -----------------------------------------------------------------------------
__global__ __launch_bounds__(128)
void gat_aggr_kernel(const float* __restrict__ h,
                     const float* __restrict__ s_in,
                     const float* __restrict__ s_out,
                     const int*   __restrict__ col,
                     const float* __restrict__ adj_vals,
                     const float* __restrict__ bias,
                     float* __restrict__ out)
{
    const int node = blockIdx.x;
    const int t    = threadIdx.x;        // feature index 0..127

    __shared__ float att[KNBR];
    __shared__ int   cols[KNBR];

    if (t < KNBR) {
        const int e = node * KNBR + t;
        const int c = col[e];
        cols[t] = c;
        float ev = s_in[node] + s_out[c];
        att[t] = (ev >= 0.f) ? ev : ALPHA * ev;   // LeakyReLU
    }
    __syncthreads();

    if (t == 0) {                         // softmax over 12 values + adj fold
        float mx = att[0];
        #pragma unroll
        for (int j = 1; j < KNBR; ++j) mx = fmaxf(mx, att[j]);
        float tmp[KNBR];
        float s = 0.f;
        #pragma unroll
        for (int j = 0; j < KNBR; ++j) { tmp[j] = __expf(att[j] - mx); s += tmp[j]; }
        const float inv = 1.f / s;
        #pragma unroll
        for (int j = 0; j < KNBR; ++j)
            att[j] = tmp[j] * inv * adj_vals[node * KNBR + j];
    }
    __syncthreads();

    float acc = 0.f;
    #pragma unroll
    for (int j = 0; j < KNBR; ++j)
        acc = fmaf(att[j], h[(size_t)cols[j] * FDIM + t], acc);

    const size_t o = (size_t)node * FDIM + t;
    out[o] = out[o] + acc + bias[t];      // act_self + act_neighbor + bias
}

// -----------------------------------------------------------------------------
extern "C" void kernel_launch(void* const* d_in, const int* in_sizes, int n_in,
                              void* d_out, int out_size, void* d_ws, size_t ws_size,
                              hipStream_t stream)
{
    const float* x        = (const float*)d_in[0];
    const float* adj_vals = (const float*)d_in[1];
    // d_in[2] = row (implicit: node*K..node*K+K-1, edges sorted by row)
    const int*   col      = (const int*)  d_in[3];
    const float* W_self   = (const float*)d_in[4];
    const float* W_ngb    = (const float*)d_in[5];
    const float* avec     = (const float*)d_in[6];
    const float* bias     = (const float*)d_in[7];
    float*       out      = (float*)d_out;

    // workspace layout: h [N*128] | s_in [N] | s_out [N] | WtS [128*128] | WtN
    float* h     = (float*)d_ws;
    float* s_in  = h + (size_t)NNODES * FDIM;
    float* s_out = s_in + NNODES;
    float* WtS   = s_out + NNODES;
    float* WtN   = WtS + FDIM * FDIM;

    transpose_w_kernel<<<(FDIM * FDIM) / 256, 256, 0, stream>>>(
        W_self, W_ngb, WtS, WtN);

    gat_gemm_kernel<<<NNODES / ROWS_PB, 256, 0, stream>>>(
        x, WtS, WtN, avec, out, h, s_in, s_out);

    gat_aggr_kernel<<<NNODES, 128, 0, stream>>>(
        h, s_in, s_out, col, adj_vals, bias, out);
}